// DGCNNSegmentation_58394375356472
// MI455X (gfx1250) — compile-verified
//
#include <hip/hip_runtime.h>

typedef __attribute__((ext_vector_type(2))) float v2f;
typedef __attribute__((ext_vector_type(8))) float v8f;

#define B_    8
#define N_    2048
#define KNN_  20
#define SLOPE_ 0.2f
#define EPS_  1e-5f

__device__ __forceinline__ float lrelu(float y) { return y > 0.f ? y : SLOPE_ * y; }

// ---------------------------------------------------------------------------
// Squared norms per point: norms[b*N+n] = sum_c x[b,c,n]^2
// Grid: (N/256, B) so the channel base pointer is block-uniform.
// ---------------------------------------------------------------------------
__global__ void norms_kernel(const float* __restrict__ x, int chStride, int chOff, int C,
                             float* __restrict__ norms) {
    const int b = blockIdx.y;
    const int n = blockIdx.x * blockDim.x + threadIdx.x;
    const float* xb = x + ((size_t)b * chStride + chOff) * (size_t)N_;
    float s = 0.f;
    for (int c = 0; c < C; ++c) {
        float v = xb[(size_t)c * N_ + n];
        s = fmaf(v, v, s);
    }
    norms[(size_t)b * N_ + n] = s;
}

// ---------------------------------------------------------------------------
// KNN: top-20 by score = 2*dot(x_n, x_m) - ||x_m||^2  (same ranking as
// -||x_n - x_m||^2; the -||x_n||^2 term is constant per query).
// Grid: (N/128, B). b is block-uniform, so candidate loads x[:,m] and
// norms[m] are uniform-address (broadcast / scalarizable); the query point
// is cached in registers.
// ---------------------------------------------------------------------------
template <int C>
__global__ void knn_kernel(const float* __restrict__ x, int chStride, int chOff,
                           const float* __restrict__ norms, int* __restrict__ idxOut) {
    const int b = blockIdx.y;
    const int n = blockIdx.x * blockDim.x + threadIdx.x;
    const float* xb = x + ((size_t)b * chStride + chOff) * (size_t)N_;
    float xn[C];
#pragma unroll
    for (int c = 0; c < C; ++c) xn[c] = xb[(size_t)c * N_ + n];
    const float* nb = norms + (size_t)b * N_;

    float best[KNN_];
    int   bidx[KNN_];
#pragma unroll
    for (int j = 0; j < KNN_; ++j) { best[j] = -__builtin_inff(); bidx[j] = 0; }

    for (int m = 0; m < N_; ++m) {
        float dot = 0.f;
#pragma unroll
        for (int c = 0; c < C; ++c) dot = fmaf(xn[c], xb[(size_t)c * N_ + m], dot);
        float score = 2.f * dot - nb[m];
        if (score > best[KNN_ - 1]) {      // rarely taken after warm-up
            float s = score; int id = m;
#pragma unroll
            for (int j = 0; j < KNN_; ++j) {
                if (s > best[j]) {
                    float tb = best[j]; int ti = bidx[j];
                    best[j] = s; bidx[j] = id;
                    s = tb; id = ti;
                }
            }
        }
    }
    int* op = idxOut + ((size_t)b * N_ + n) * KNN_;
#pragma unroll
    for (int j = 0; j < KNN_; ++j) op[j] = bidx[j];
}

// ---------------------------------------------------------------------------
// EdgeConv: out[b, oc, n] = max_k lrelu(BN(W @ [center; nbr-center]))
// One block (4 waves) per (n, b). Edge matrix (CIN2 x 20, padded to 32 cols)
// staged in LDS; each wave computes 16-output-channel tiles with
// v_wmma_f32_16x16x4_f32 over two 16-column tiles (k=0..15, 16..19).
// Max-over-k = segmented 16-lane shuffle reduction, batched over all 8
// accumulator rows per round so the ds_bpermutes overlap.
// ---------------------------------------------------------------------------
template <int CIN2>
__global__ void edgeconv_kernel(const float* __restrict__ x, int inStride, int inOff,
                                const int* __restrict__ idx,
                                const float* __restrict__ W,
                                const float* __restrict__ bn_g, const float* __restrict__ bn_b,
                                const float* __restrict__ bn_m, const float* __restrict__ bn_v,
                                float* __restrict__ out, int outStride, int outOff, int Cout) {
    constexpr int C = CIN2 / 2;
    __shared__ float edge[CIN2 * 32];
    __shared__ float cx[C];
    __shared__ int   nidx[32];

    const int n = blockIdx.x, b = blockIdx.y;
    const int tid = threadIdx.x;
    const float* xb = x + ((size_t)b * inStride + inOff) * (size_t)N_;

    if (tid < 32) nidx[tid] = (tid < KNN_) ? idx[((size_t)b * N_ + n) * KNN_ + tid] : 0;
    if (tid < C)  cx[tid] = xb[(size_t)tid * N_ + n];
    __syncthreads();

    for (int e = tid; e < CIN2 * 32; e += blockDim.x) {
        int c = e >> 5, col = e & 31;
        float val = 0.f;
        if (col < KNN_) {
            if (c < C) val = cx[c];
            else       val = xb[(size_t)(c - C) * N_ + nidx[col]] - cx[c - C];
        }
        edge[e] = val;
    }
    __syncthreads();

    const int lane = tid & 31;
    const int wave = tid >> 5;
    const int row  = lane & 15;
    const int koff = (lane >> 4) << 1;   // lanes 0-15 -> K rows 0,1 ; lanes 16-31 -> K rows 2,3
    const int nMT  = Cout >> 4;

    for (int mt = wave; mt < nMT; mt += 4) {
        const float* Wr = W + (size_t)(mt * 16 + row) * CIN2 + koff;
        v8f acc0 = {0.f, 0.f, 0.f, 0.f, 0.f, 0.f, 0.f, 0.f};
        v8f acc1 = {0.f, 0.f, 0.f, 0.f, 0.f, 0.f, 0.f, 0.f};
        for (int k0 = 0; k0 < CIN2; k0 += 4) {
            v2f a = *(const v2f*)(Wr + k0);
            v2f b0, b1;
            b0.x = edge[(k0 + koff)     * 32 + row];
            b0.y = edge[(k0 + koff + 1) * 32 + row];
            b1.x = edge[(k0 + koff)     * 32 + 16 + row];
            b1.y = edge[(k0 + koff + 1) * 32 + 16 + row];
            acc0 = __builtin_amdgcn_wmma_f32_16x16x4_f32(false, a, false, b0, (short)0, acc0, false, false);
            acc1 = __builtin_amdgcn_wmma_f32_16x16x4_f32(false, a, false, b1, (short)0, acc1, false, false);
        }
        // Epilogue: BN + leaky relu, then max over neighbor columns.
        // D layout: VGPR r, lanes 0-15 -> (M=r, col=lane); lanes 16-31 -> (M=8+r, col=lane-16).
        const int  chBase    = mt * 16 + ((lane >> 4) << 3);
        const bool validCol1 = (row < (KNN_ - 16));   // acc1 columns 16..19 valid
        float mx[8];
#pragma unroll
        for (int r = 0; r < 8; ++r) {
            int ch = chBase + r;
            float sc = bn_g[ch] * rsqrtf(bn_v[ch] + EPS_);
            float tb = bn_b[ch] - bn_m[ch] * sc;
            float y0 = lrelu(acc0[r] * sc + tb);
            float y1 = lrelu(acc1[r] * sc + tb);
            mx[r] = fmaxf(y0, validCol1 ? y1 : -__builtin_inff());
        }
        // Segmented (16-lane) max reduction, round-major so the 8 bpermutes
        // per round are independent and can be in flight together.
#pragma unroll
        for (int off = 8; off > 0; off >>= 1) {
            float other[8];
#pragma unroll
            for (int r = 0; r < 8; ++r) other[r] = __shfl_xor(mx[r], off, 16);
#pragma unroll
            for (int r = 0; r < 8; ++r) mx[r] = fmaxf(mx[r], other[r]);
        }
        if (row == 0) {
#pragma unroll
            for (int r = 0; r < 8; ++r)
                out[((size_t)b * outStride + outOff + chBase + r) * (size_t)N_ + n] = mx[r];
        }
    }
}

// ---------------------------------------------------------------------------
// Pointwise conv + BN + leaky relu.  y[b,oc,n] = lrelu(BN(sum_c W[oc,c]*h[c,n]))
// Channels c >= Cin1 come from the per-batch broadcast vector gvec (dec1's
// concat with the global max), so the 1024-channel input is never materialized.
// One block (4 waves) per (point-tile-of-16, b); B tile staged in LDS.
// ---------------------------------------------------------------------------
template <int CTOT>
__global__ void pconv_kernel(const float* __restrict__ x, int Cin1, int xStride, int xOff,
                             const float* __restrict__ gvec,
                             const float* __restrict__ W,
                             const float* __restrict__ bn_g, const float* __restrict__ bn_b,
                             const float* __restrict__ bn_m, const float* __restrict__ bn_v,
                             float* __restrict__ out, int outStride, int outOff, int Cout) {
    __shared__ float bt[CTOT * 16];
    const int nt = blockIdx.x, b = blockIdx.y;
    const int tid = threadIdx.x;

    for (int e = tid; e < CTOT * 16; e += blockDim.x) {
        int c = e >> 4, col = e & 15;
        float val;
        if (c < Cin1) val = x[((size_t)b * xStride + xOff + c) * (size_t)N_ + nt * 16 + col];
        else          val = gvec[(size_t)b * (CTOT - Cin1) + (c - Cin1)];
        bt[e] = val;
    }
    __syncthreads();

    const int lane = tid & 31;
    const int wave = tid >> 5;
    const int row  = lane & 15;
    const int koff = (lane >> 4) << 1;
    const int nMT  = Cout >> 4;

    for (int mt = wave; mt < nMT; mt += 4) {
        const float* Wr = W + (size_t)(mt * 16 + row) * CTOT + koff;
        v8f acc = {0.f, 0.f, 0.f, 0.f, 0.f, 0.f, 0.f, 0.f};
        for (int k0 = 0; k0 < CTOT; k0 += 4) {
            v2f a = *(const v2f*)(Wr + k0);
            v2f bb;
            bb.x = bt[(k0 + koff)     * 16 + row];
            bb.y = bt[(k0 + koff + 1) * 16 + row];
            acc = __builtin_amdgcn_wmma_f32_16x16x4_f32(false, a, false, bb, (short)0, acc, false, false);
        }
        const int chBase = mt * 16 + ((lane >> 4) << 3);
        const int n = nt * 16 + row;
#pragma unroll
        for (int r = 0; r < 8; ++r) {
            int ch = chBase + r;
            float sc = bn_g[ch] * rsqrtf(bn_v[ch] + EPS_);
            float tb = bn_b[ch] - bn_m[ch] * sc;
            float y = lrelu(acc[r] * sc + tb);
            out[((size_t)b * outStride + outOff + ch) * (size_t)N_ + n] = y;
        }
    }
}

// ---------------------------------------------------------------------------
// Global max over points: glob[b,c] = max_n local[b,c,n]
// ---------------------------------------------------------------------------
__global__ void gmax_kernel(const float* __restrict__ local, float* __restrict__ glob) {
    const int c = blockIdx.x, b = blockIdx.y;
    const float* p = local + ((size_t)b * 512 + c) * (size_t)N_;
    float mx = -__builtin_inff();
    for (int n = threadIdx.x; n < N_; n += blockDim.x) mx = fmaxf(mx, p[n]);
    __shared__ float red[256];
    red[threadIdx.x] = mx;
    __syncthreads();
    for (int s = 128; s > 0; s >>= 1) {
        if (threadIdx.x < s) red[threadIdx.x] = fmaxf(red[threadIdx.x], red[threadIdx.x + s]);
        __syncthreads();
    }
    if (threadIdx.x == 0) glob[(size_t)b * 512 + c] = red[0];
}

// ---------------------------------------------------------------------------
// Final 13x256 head with bias (small M -> plain VALU kernel).
// Grid: (N/256, B); W/bias loads are uniform (scalar path).
// ---------------------------------------------------------------------------
__global__ void dec3_kernel(const float* __restrict__ h, const float* __restrict__ W,
                            const float* __restrict__ bias, float* __restrict__ out) {
    const int b = blockIdx.y;
    const int n = blockIdx.x * blockDim.x + threadIdx.x;
    const float* hb = h + (size_t)b * 256 * N_ + n;
    float acc[13];
#pragma unroll
    for (int o = 0; o < 13; ++o) acc[o] = bias[o];
    for (int c = 0; c < 256; ++c) {
        float xv = hb[(size_t)c * N_];
#pragma unroll
        for (int o = 0; o < 13; ++o) acc[o] = fmaf(W[o * 256 + c], xv, acc[o]);
    }
#pragma unroll
    for (int o = 0; o < 13; ++o) out[((size_t)b * 13 + o) * (size_t)N_ + n] = acc[o];
}

// ---------------------------------------------------------------------------
extern "C" void kernel_launch(void* const* d_in, const int* in_sizes, int n_in,
                              void* d_out, int out_size, void* d_ws, size_t ws_size,
                              hipStream_t stream) {
    (void)in_sizes; (void)n_in; (void)out_size; (void)ws_size;

    const float* x   = (const float*)d_in[0];
    const float* W1  = (const float*)d_in[1];
    const float *g1 = (const float*)d_in[2], *b1 = (const float*)d_in[3],
                *m1 = (const float*)d_in[4], *v1 = (const float*)d_in[5];
    const float* W2  = (const float*)d_in[6];
    const float *g2 = (const float*)d_in[7], *b2 = (const float*)d_in[8],
                *m2 = (const float*)d_in[9], *v2 = (const float*)d_in[10];
    const float* W3  = (const float*)d_in[11];
    const float *g3 = (const float*)d_in[12], *b3 = (const float*)d_in[13],
                *m3 = (const float*)d_in[14], *v3 = (const float*)d_in[15];
    const float* W4  = (const float*)d_in[16];
    const float *g4 = (const float*)d_in[17], *b4 = (const float*)d_in[18],
                *m4 = (const float*)d_in[19], *v4 = (const float*)d_in[20];
    const float* Wf  = (const float*)d_in[21];
    const float *gf = (const float*)d_in[22], *bf = (const float*)d_in[23],
                *mf = (const float*)d_in[24], *vf = (const float*)d_in[25];
    const float* Wd1 = (const float*)d_in[26];
    const float *gd1 = (const float*)d_in[27], *bd1 = (const float*)d_in[28],
                *md1 = (const float*)d_in[29], *vd1 = (const float*)d_in[30];
    const float* Wd2 = (const float*)d_in[31];
    const float *gd2 = (const float*)d_in[32], *bd2 = (const float*)d_in[33],
                *md2 = (const float*)d_in[34], *vd2 = (const float*)d_in[35];
    const float* Wd3 = (const float*)d_in[36];
    const float* bd3 = (const float*)d_in[37];

    // Workspace layout (bytes):
    //   [0, 32MB)        cat  (B,512,N): x1 ch0-63, x2 ch64-127, x3 ch128-255, x4 ch256-511
    //                    (reused as dec1 output h1 after fuse)
    //   [32MB, 64MB)     local (B,512,N) fuse output (reused as dec2 output h2)
    //   [64MB, ...)      idx (B,N,20) int, norms (B,N), glob (B,512)
    char* ws = (char*)d_ws;
    float* cat   = (float*)(ws);
    float* loc   = (float*)(ws + (size_t)33554432);
    int*   idx   = (int*)  (ws + (size_t)67108864);
    float* norms = (float*)(ws + (size_t)67108864 + 1310720);
    float* glob  = (float*)(ws + (size_t)67108864 + 1310720 + 65536);

    dim3 gNRM(N_ / 256, B_);
    dim3 gKNN(N_ / 128, B_);
    dim3 gEC(N_, B_);
    dim3 gPC(N_ / 16, B_);

    // ---- ec1: x (C=6) -> x1 (64ch, cat off 0)
    norms_kernel<<<gNRM, 256, 0, stream>>>(x, 6, 0, 6, norms);
    knn_kernel<6><<<gKNN, 128, 0, stream>>>(x, 6, 0, norms, idx);
    edgeconv_kernel<12><<<gEC, 128, 0, stream>>>(x, 6, 0, idx, W1, g1, b1, m1, v1,
                                                 cat, 512, 0, 64);
    // ---- ec2: x1 -> x2 (64ch, cat off 64)
    norms_kernel<<<gNRM, 256, 0, stream>>>(cat, 512, 0, 64, norms);
    knn_kernel<64><<<gKNN, 128, 0, stream>>>(cat, 512, 0, norms, idx);
    edgeconv_kernel<128><<<gEC, 128, 0, stream>>>(cat, 512, 0, idx, W2, g2, b2, m2, v2,
                                                  cat, 512, 64, 64);
    // ---- ec3: x2 -> x3 (128ch, cat off 128)
    norms_kernel<<<gNRM, 256, 0, stream>>>(cat, 512, 64, 64, norms);
    knn_kernel<64><<<gKNN, 128, 0, stream>>>(cat, 512, 64, norms, idx);
    edgeconv_kernel<128><<<gEC, 128, 0, stream>>>(cat, 512, 64, idx, W3, g3, b3, m3, v3,
                                                  cat, 512, 128, 128);
    // ---- ec4: x3 -> x4 (256ch, cat off 256)
    norms_kernel<<<gNRM, 256, 0, stream>>>(cat, 512, 128, 128, norms);
    knn_kernel<128><<<gKNN, 128, 0, stream>>>(cat, 512, 128, norms, idx);
    edgeconv_kernel<256><<<gEC, 128, 0, stream>>>(cat, 512, 128, idx, W4, g4, b4, m4, v4,
                                                  cat, 512, 256, 256);
    // ---- fuse: cat (512) -> local (512)
    pconv_kernel<512><<<gPC, 128, 0, stream>>>(cat, 512, 512, 0, (const float*)nullptr,
                                               Wf, gf, bf, mf, vf, loc, 512, 0, 512);
    // ---- global max
    gmax_kernel<<<dim3(512, B_), 256, 0, stream>>>(loc, glob);
    // ---- dec1: [local ; glob] (1024) -> h1 (512, reuses cat region)
    pconv_kernel<1024><<<gPC, 128, 0, stream>>>(loc, 512, 512, 0, glob,
                                                Wd1, gd1, bd1, md1, vd1, cat, 512, 0, 512);
    // ---- dec2: h1 (512) -> h2 (256, reuses local region)
    pconv_kernel<512><<<gPC, 128, 0, stream>>>(cat, 512, 512, 0, (const float*)nullptr,
                                               Wd2, gd2, bd2, md2, vd2, loc, 256, 0, 256);
    // ---- dec3 head: h2 -> logits
    dec3_kernel<<<gNRM, 256, 0, stream>>>(loc, Wd3, bd3, (float*)d_out);
}